// RuleConvolutionLayer_81965155877382
// MI455X (gfx1250) — compile-verified
//
#include <hip/hip_runtime.h>
#include <stdint.h>

// Rule-convolution message passing:
//   out[v] = b[label[v]] + sum_{e:(v,w,p)} W[(label[v]*L + label[w])*P + p] * x[w]
// Memory-bound gather/scatter. Edge index streams are pipelined into LDS with
// the CDNA5 async copy engine (global_load_async_to_lds_b128 / ASYNCcnt); each
// lane consumes exactly its own staged 16B chunks, so no block barriers are
// needed. Accumulation uses no-return global_atomic_add_f32 (STOREcnt).

#define NBLOCK 256           // 8 wave32 per block
#define EPT 4                // edges per thread per tile (one B128 per stream)
#define TILE (NBLOCK * EPT)  // 1024 edges per tile
#define MAX_GRID 4096

// CDNA5 async global -> LDS copy, 16 bytes per lane (GVS addressing).
__device__ __forceinline__ void async_ld_b128(uint32_t lds_addr, uint32_t voff,
                                              const void* base) {
    asm volatile("global_load_async_to_lds_b128 %0, %1, %2"
                 :
                 : "v"(lds_addr), "v"(voff), "s"(base)
                 : "memory");
}

// Fire-and-forget fp32 atomic add at device scope (no return value).
__device__ __forceinline__ void atomic_add_f32_dev(const float* base,
                                                   uint32_t byte_off, float val) {
    asm volatile("global_atomic_add_f32 %0, %1, %2 scope:SCOPE_DEV"
                 :
                 : "v"(byte_off), "v"(val), "s"(base)
                 : "memory");
}

__global__ void __launch_bounds__(NBLOCK)
bias_init_kernel(const float* __restrict__ b, const int* __restrict__ label,
                 float* __restrict__ out, int n) {
    int i = blockIdx.x * NBLOCK + threadIdx.x;
    if (i < n) out[i] = b[label[i]];
}

__global__ void __launch_bounds__(NBLOCK)
edge_msg_kernel(const float* __restrict__ x, const float* __restrict__ W,
                const int* __restrict__ label,
                const int* __restrict__ esrc, const int* __restrict__ edst,
                const int* __restrict__ eprop, float* __restrict__ out,
                int n_edges, int n_tiles, int n_labels, int n_props) {
    __shared__ __align__(16) int sm[2][3][TILE];
    const int tid = threadIdx.x;

    // Per-buffer LDS byte addresses of this lane's private 16B slot per stream.
    uint32_t l_s[2], l_d[2], l_p[2];
#pragma unroll
    for (int q = 0; q < 2; ++q) {
        l_s[q] = (uint32_t)(uintptr_t)&sm[q][0][tid * 4];
        l_d[q] = (uint32_t)(uintptr_t)&sm[q][1][tid * 4];
        l_p[q] = (uint32_t)(uintptr_t)&sm[q][2][tid * 4];
    }

    int t = blockIdx.x;
    int pb = 0;
    if (t < n_tiles) {
        uint32_t voff = ((uint32_t)t * TILE + (uint32_t)tid * 4) * 4u;
        async_ld_b128(l_s[0], voff, esrc);
        async_ld_b128(l_d[0], voff, edst);
        async_ld_b128(l_p[0], voff, eprop);
    }

    for (; t < n_tiles; t += gridDim.x) {
        int tn = t + (int)gridDim.x;
        if (tn < n_tiles) {
            // Prefetch next tile into the other buffer; then wait until only
            // those 3 newest async ops remain in flight (async loads retire
            // in order -> tile t's chunks are resident in LDS).
            uint32_t voff = ((uint32_t)tn * TILE + (uint32_t)tid * 4) * 4u;
            async_ld_b128(l_s[pb ^ 1], voff, esrc);
            async_ld_b128(l_d[pb ^ 1], voff, edst);
            async_ld_b128(l_p[pb ^ 1], voff, eprop);
            asm volatile("s_wait_asynccnt 3" ::: "memory");
        } else {
            asm volatile("s_wait_asynccnt 0" ::: "memory");
        }

        // Each lane reads back exactly the chunks it staged: 3x ds_load_b128,
        // no cross-thread sharing, no barrier required.
        const int4 vs = *(const int4*)&sm[pb][0][tid * 4];
        const int4 vd = *(const int4*)&sm[pb][1][tid * 4];
        const int4 vp = *(const int4*)&sm[pb][2][tid * 4];

        const int v[EPT] = {vs.x, vs.y, vs.z, vs.w};
        const int w[EPT] = {vd.x, vd.y, vd.z, vd.w};
        const int p[EPT] = {vp.x, vp.y, vp.z, vp.w};
#pragma unroll
        for (int k = 0; k < EPT; ++k) {
            int lv = label[v[k]];  // L2-resident gathers
            int lw = label[w[k]];
            float msg = W[(lv * n_labels + lw) * n_props + p[k]] * x[w[k]];
            atomic_add_f32_dev(out, (uint32_t)v[k] * 4u, msg);
        }
        pb ^= 1;
    }

    // Remainder edges (n_edges not a multiple of TILE): direct guarded path.
    int rem_start = n_tiles * TILE;
    for (int e = rem_start + blockIdx.x * NBLOCK + tid; e < n_edges;
         e += gridDim.x * NBLOCK) {
        int v = esrc[e];
        int w = edst[e];
        int p = eprop[e];
        int lv = label[v];
        int lw = label[w];
        float msg = W[(lv * n_labels + lw) * n_props + p] * x[w];
        atomic_add_f32_dev(out, (uint32_t)v * 4u, msg);
    }
}

extern "C" void kernel_launch(void* const* d_in, const int* in_sizes, int n_in,
                              void* d_out, int out_size, void* d_ws, size_t ws_size,
                              hipStream_t stream) {
    const float* x     = (const float*)d_in[0];
    const float* W     = (const float*)d_in[1];
    const float* b     = (const float*)d_in[2];
    const int*   label = (const int*)d_in[3];
    const int*   esrc  = (const int*)d_in[4];
    const int*   edst  = (const int*)d_in[5];
    const int*   eprop = (const int*)d_in[6];
    float* out = (float*)d_out;

    int n_nodes  = in_sizes[0];
    int n_edges  = in_sizes[4];
    int n_labels = in_sizes[2];
    int n_props  = in_sizes[1] / (n_labels * n_labels);
    int n_tiles  = n_edges / TILE;

    // out[v] = b[label[v]] (also initializes the poisoned output buffer)
    bias_init_kernel<<<(n_nodes + NBLOCK - 1) / NBLOCK, NBLOCK, 0, stream>>>(
        b, label, out, n_nodes);

    int grid = n_tiles < MAX_GRID ? (n_tiles > 0 ? n_tiles : 1) : MAX_GRID;
    edge_msg_kernel<<<grid, NBLOCK, 0, stream>>>(
        x, W, label, esrc, edst, eprop, out, n_edges, n_tiles, n_labels, n_props);
}